// SparseTemporalMemory_16741782520507
// MI455X (gfx1250) — compile-verified
//
#include <hip/hip_runtime.h>
#include <math.h>
#include <stdint.h>

#define B_    8
#define D_    256
#define MEM_  131072
#define CW_   64
#define R_    4
#define K_    4
#define KL_   4
#define C_    25               // R*K + 2*KL + 1
#define NCH   64               // chunks per batch in the big scan
#define CHROWS (MEM_ / NCH)    // 2048 rows per block
#define DELTA_ 0.005f
#define EPS_   1e-6f
#define LPAD  68               // 64 + 4 dword row padding (bank-conflict-free column reads)

typedef __attribute__((ext_vector_type(2))) float v2f;
typedef __attribute__((ext_vector_type(8))) float v8f;

#if defined(__HIP_DEVICE_COMPILE__) && defined(__gfx1250__) && __has_builtin(__builtin_amdgcn_wmma_f32_16x16x4_f32)
#define USE_WMMA 1
#else
#define USE_WMMA 0
#endif

// async global->LDS staging (gfx1250 ASYNCcnt path); inline asm per CDNA5_HIP.md guidance
#if USE_WMMA
#define USE_ASYNC 1
#else
#define USE_ASYNC 0
#endif

// ---------------- workspace layout ----------------
struct WS {
  float rq[B_][R_][CW_];       // read_query (raw)
  float vmnew[B_][C_][CW_];    // updated visible_memory rows (the scatter payload)
  float nsc[B_][C_][R_];       // exact scores 2*q.v - |v|^2 for overridden rows
  float uover[B_][C_];         // updated usage values at read positions
  int   fpbp[B_][8];           // fp[4], bp[4]
  int   knn[B_][R_ * K_];
  unsigned long long amin;     // packed (mono(usage)<<32)|idx, batch 0
  unsigned long long btop[B_][NCH][R_][4];  // per-block top-4 packed keys
};

__device__ __forceinline__ unsigned int fmono(float f) {
  unsigned int u = __float_as_uint(f);
  return (u & 0x80000000u) ? ~u : (u | 0x80000000u);  // order-preserving for signed floats
}
__device__ __forceinline__ unsigned int umin_u(unsigned int a, unsigned int b) { return a < b ? a : b; }

// larger key = better: (mono(value) << 32) | ~index  -> value desc, index asc on ties
__device__ __forceinline__ void ins4(unsigned long long k, unsigned long long t[4]) {
  if (k > t[0]) { t[3]=t[2]; t[2]=t[1]; t[1]=t[0]; t[0]=k; }
  else if (k > t[1]) { t[3]=t[2]; t[2]=t[1]; t[1]=k; }
  else if (k > t[2]) { t[3]=t[2]; t[2]=k; }
  else if (k > t[3]) { t[3]=k; }
}

#if USE_ASYNC
// Copy one 16x64-f32 tile (4 KB, 16 consecutive memory rows) into LDS with a
// 4-dword pad after each row. 8 x global_load_async_to_lds_b128: each instr
// moves 32 lanes x 16B = 512B (= 2 rows). Tracked by ASYNCcnt.
__device__ __forceinline__ void issue_tile_async(const float* gtile, const float* lbase,
                                                 int half, int nlo, int lane) {
  unsigned lofs = (unsigned)(uintptr_t)lbase + (unsigned)((half * LPAD + 4 * nlo) * 4);
  unsigned long long g = (unsigned long long)(uintptr_t)gtile + (unsigned)(16 * lane);
  #pragma unroll
  for (int i = 0; i < 8; ++i) {
    asm volatile("global_load_async_to_lds_b128 %0, %1, off"
                 :: "v"(lofs), "v"(g) : "memory");
    lofs += 2u * LPAD * 4u;   // two (padded) rows per step
    g += 512ull;              // two rows in global memory
  }
}
#endif

// ---------------- kernel 1: projections + state updates + fp/bp ----------------
__global__ __launch_bounds__(256) void prep_kernel(
    const float* __restrict__ xi,   const float* __restrict__ vmem,
    const float* __restrict__ link, const float* __restrict__ rev,
    const float* __restrict__ prec, const float* __restrict__ rwt,
    const float* __restrict__ wwt,  const float* __restrict__ usage,
    const int*   __restrict__ rp,
    const float* __restrict__ W_rq, const float* __restrict__ b_rq,
    const float* __restrict__ W_wv, const float* __restrict__ b_wv,
    const float* __restrict__ W_ig, const float* __restrict__ b_ig,
    const float* __restrict__ W_wg, const float* __restrict__ b_wg,
    WS* __restrict__ ws) {
  const int b = blockIdx.x;
  const int t = threadIdx.x;
  __shared__ float xs[D_];
  __shared__ float rqs[R_ * CW_];
  __shared__ float wvs[CW_];
  __shared__ float igs[C_];
  __shared__ float wgs;
  __shared__ float relv[C_], iminv[C_], wwns[C_];
  __shared__ float relmin;
  __shared__ float vmn[C_][CW_];

  xs[t] = xi[b * D_ + t];
  __syncthreads();
  { // read_query: 256 outputs, one per thread
    float acc = b_rq[t];
    for (int d = 0; d < D_; ++d) acc += xs[d] * W_rq[d * (R_ * CW_) + t];
    rqs[t] = acc;
    ((float*)ws->rq[b])[t] = acc;
  }
  if (t < CW_) { float a = b_wv[t]; for (int d = 0; d < D_; ++d) a += xs[d] * W_wv[d * CW_ + t]; wvs[t] = a; }
  if (t < C_)  { float a = b_ig[t]; for (int d = 0; d < D_; ++d) a += xs[d] * W_ig[d * C_ + t]; igs[t] = 1.f / (1.f + expf(-a)); }
  if (t == 0)  { float a = b_wg[0]; for (int d = 0; d < D_; ++d) a += xs[d] * W_wg[d]; wgs = 1.f / (1.f + expf(-a)); }
  if (t < C_)  relv[t] = usage[(size_t)b * MEM_ + rp[b * C_ + t]];
  __syncthreads();
  if (t == 0) { float mn = relv[0]; for (int j = 1; j < C_; ++j) mn = fminf(mn, relv[j]); relmin = mn; }
  __syncthreads();
  if (t < C_) {
    int p = rp[b * C_ + t];
    float rv = rwt[(size_t)b * MEM_ + p];
    float wv = wwt[(size_t)b * MEM_ + p];
    float u  = (rv + wv > DELTA_) ? 1.f : 0.f;
    float im = (relv[t] == relmin) ? 1.f : 0.f;
    iminv[t] = im;
    ws->uover[b][t] = u * (1.f - relv[t]) + (1.f - u) * relv[t];   // TIMESTEP=1
    wwns[t] = wgs * (igs[t] * rv + (1.f - igs[t]) * im);
  }
  __syncthreads();
  for (int i = t; i < C_ * CW_; i += 256) {  // updated visible_memory rows
    int j = i / CW_, wc = i % CW_;
    float v = vmem[((size_t)b * C_ + j) * CW_ + wc] * (1.f - iminv[j]) + wwns[j] * wvs[wc];
    vmn[j][wc] = v;
    ws->vmnew[b][j][wc] = v;
  }
  __syncthreads();
  if (t < C_ * R_) {  // exact scores for the 25 overridden memory rows
    int j = t / R_, r = t % R_;
    float dp = 0.f, sq = 0.f;
    for (int wc = 0; wc < CW_; ++wc) { float v = vmn[j][wc]; dp += rqs[r * CW_ + wc] * v; sq += v * v; }
    ws->nsc[b][j][r] = 2.f * dp - sq;
  }
  if (t == 0) {
    // Ieye = eye(MEM, 8) => only the 8x8 diagonal corner of link/rev survives.
    int rl[C_];  for (int j = 0; j < C_; ++j) rl[j] = rp[b * C_ + j];
    int trp[8];  for (int k = 0; k < 8; ++k) trp[k] = rl[R_ * K_ + k];
    float fw[12], bw[12];
    for (int m = 0; m < 8; ++m) {
      float wwf = wwt[(size_t)b * MEM_ + m];
      for (int j = 0; j < C_; ++j) if (rl[j] == m) wwf = wwns[j];          // write_weights scatter (last wins)
      float trwm = rwt[(size_t)b * MEM_ + trp[m]];
      float twwm = wwt[(size_t)b * MEM_ + trp[m]];
      for (int j = 0; j < C_; ++j) if (rl[j] == trp[m]) twwm = wwns[j];    // tww = gather(new ww, trp)
      float pden = 0.f;
      for (int k = 0; k < 8; ++k) if (trp[k] == m) pden = prec[b * 8 + k]; // prec_dense scatter
      float ld = link[(((size_t)b * MEM_) + m) * 8 + m];
      float rd = rev [(((size_t)b * MEM_) + m) * 8 + m];
      fw[m] = ((1.f - wwf)  * ld + wwf  * prec[b * 8 + m]) * trwm;
      bw[m] = ((1.f - twwm) * rd + twwm * pden)            * trwm;
    }
    for (int m = 8; m < 12; ++m) { fw[m] = 0.f; bw[m] = 0.f; }  // virtual zero entries (indices 8..11)
    bool used[12];
    for (int i = 0; i < 12; ++i) used[i] = false;
    for (int e = 0; e < 4; ++e) { int bs = -1; float bv = 0.f;
      for (int m = 0; m < 12; ++m) if (!used[m] && (bs < 0 || fw[m] > bv)) { bs = m; bv = fw[m]; }
      used[bs] = true; ws->fpbp[b][e] = bs; }
    for (int i = 0; i < 12; ++i) used[i] = false;
    for (int e = 0; e < 4; ++e) { int bs = -1; float bv = 0.f;
      for (int m = 0; m < 12; ++m) if (!used[m] && (bs < 0 || bw[m] > bv)) { bs = m; bv = bw[m]; }
      used[bs] = true; ws->fpbp[b][4 + e] = bs; }
    if (b == 0) ws->amin = ~0ULL;  // init for argmin kernel (stream-ordered)
  }
}

// ---------------- kernel 2: argmin of updated usage, batch 0 ----------------
__global__ __launch_bounds__(256) void argmin_kernel(const float* __restrict__ usage,
                                                     const int* __restrict__ rp,
                                                     WS* __restrict__ ws) {
  __shared__ int rps[C_];
  __shared__ unsigned long long red[256];
  if (threadIdx.x < C_) rps[threadIdx.x] = rp[threadIdx.x];  // batch 0
  __syncthreads();
  unsigned long long best = ~0ULL;
  const int tid = blockIdx.x * 256 + threadIdx.x;
  const int stride = gridDim.x * 256;
  for (int m = tid; m < MEM_; m += stride) {
    bool ov = false;
    #pragma unroll
    for (int j = 0; j < C_; ++j) ov = ov || (rps[j] == m);
    if (ov) continue;  // overridden entries handled below with updated values
    unsigned long long key = ((unsigned long long)fmono(usage[m]) << 32) | (unsigned int)m;
    if (key < best) best = key;
  }
  if (tid == 0) {
    for (int j = 0; j < C_; ++j) {
      bool last = true;
      for (int j2 = j + 1; j2 < C_; ++j2) if (rps[j2] == rps[j]) { last = false; break; }
      if (!last) continue;
      unsigned long long key = ((unsigned long long)fmono(ws->uover[0][j]) << 32) | (unsigned int)rps[j];
      if (key < best) best = key;
    }
  }
  red[threadIdx.x] = best;
  __syncthreads();
  for (int s = 128; s > 0; s >>= 1) {
    if (threadIdx.x < s) { if (red[threadIdx.x + s] < red[threadIdx.x]) red[threadIdx.x] = red[threadIdx.x + s]; }
    __syncthreads();
  }
  if (threadIdx.x == 0) atomicMin(&ws->amin, red[0]);
}

// ---------------- kernel 3: the 256 MB stream — async->LDS + WMMA dots + fused mem_sq + top-4 ----------------
__global__ __launch_bounds__(128) void topk_kernel(const float* __restrict__ mem,
                                                   const int* __restrict__ rp,
                                                   WS* __restrict__ ws) {
  const int b     = blockIdx.y;
  const int chunk = blockIdx.x;
  const int wave  = threadIdx.x >> 5;
  const int lane  = threadIdx.x & 31;
  const int nlo   = lane & 15;
  const int half  = lane >> 4;

  __shared__ int   rps[C_];
  __shared__ float nscs[C_][R_];
#if USE_ASYNC
  __shared__ __attribute__((aligned(16))) float ltile[4][2][16 * LPAD];  // per-wave double buffer
#endif
#if !USE_WMMA
  __shared__ float qs[R_][CW_];
#endif
  if (threadIdx.x < C_) rps[threadIdx.x] = rp[b * C_ + threadIdx.x];
  if (threadIdx.x < C_ * R_) ((float*)nscs)[threadIdx.x] = ((const float*)ws->nsc[b])[threadIdx.x];
#if !USE_WMMA
  for (int i = threadIdx.x; i < R_ * CW_; i += 128) ((float*)qs)[i] = ((const float*)ws->rq[b])[i];
#endif
  __syncthreads();

#if USE_WMMA
  // A fragment (16x4 f32, rows 4..15 zero): lanes 0-15 hold M=lane, K={w0,w0+1};
  // lanes 16-31 hold M=lane-16, K={w0+2,w0+3}. 2x folded into A.
  float A0[16], A1[16];
  #pragma unroll
  for (int t = 0; t < 16; ++t) {
    const int w = 4 * t + 2 * half;
    float q0 = 0.f, q1 = 0.f;
    if (nlo < R_) { q0 = 2.f * ws->rq[b][nlo][w]; q1 = 2.f * ws->rq[b][nlo][w + 1]; }
    A0[t] = q0; A1[t] = q1;
  }
#endif

  unsigned long long top[R_][4];
  #pragma unroll
  for (int r = 0; r < R_; ++r)
    #pragma unroll
    for (int e = 0; e < 4; ++e) top[r][e] = 0ULL;  // decodes worse than any real score

  const float* __restrict__ base = mem + (size_t)b * MEM_ * CW_;

#if USE_ASYNC
  // ---- double-buffered pipeline: tile i+1 in flight while tile i feeds WMMA ----
  const int tileStart = chunk * CHROWS + wave * 16;   // this wave's first row
  const int NT = CHROWS / 64;                         // tiles per wave (stride 64 rows)
  const float* lbuf0 = &ltile[wave][0][0];
  const float* lbuf1 = &ltile[wave][1][0];
  issue_tile_async(base + (size_t)tileStart * CW_, lbuf0, half, nlo, lane);
  for (int it = 0; it < NT; ++it) {
    const int m0 = tileStart + it * 64;
    const int m  = m0 + nlo;
    const float* lb = (it & 1) ? lbuf1 : lbuf0;
    const float* nb = (it & 1) ? lbuf0 : lbuf1;
    if (it + 1 < NT) {
      issue_tile_async(base + (size_t)(m0 + 64) * CW_, nb, half, nlo, lane);
      asm volatile("s_wait_asynccnt 0x8" ::: "memory");   // current tile resident (in-order)
    } else {
      asm volatile("s_wait_asynccnt 0x0" ::: "memory");
    }
    const float* rowl = lb + nlo * LPAD;                  // this lane's padded row
    float sq = 0.f;
    v8f acc = {0.f, 0.f, 0.f, 0.f, 0.f, 0.f, 0.f, 0.f};
    #pragma unroll
    for (int t = 0; t < 16; ++t) {
      const int w = 4 * t + 2 * half;
      float b0 = rowl[w], b1 = rowl[w + 1];               // ds_load_b64, conflict-free banks
      v2f av; av.x = A0[t]; av.y = A1[t];
      v2f bv; bv.x = b0;    bv.y = b1;
      acc = __builtin_amdgcn_wmma_f32_16x16x4_f32(false, av, false, bv, (short)0, acc, false, false);
      sq += b0 * b0 + b1 * b1;
    }
    float fsq = sq + __shfl_xor(sq, 16, 32);              // full |row|^2 (halves combined)
    if (half == 0) {
      float s[4] = { acc[0] - fsq, acc[1] - fsq, acc[2] - fsq, acc[3] - fsq };
      int ov = -1;
      #pragma unroll
      for (int j = 0; j < C_; ++j) if (rps[j] == m) ov = j;  // last occurrence wins
      if (ov >= 0) { s[0] = nscs[ov][0]; s[1] = nscs[ov][1]; s[2] = nscs[ov][2]; s[3] = nscs[ov][3]; }
      #pragma unroll
      for (int r = 0; r < R_; ++r) {
        unsigned long long key =
            ((unsigned long long)fmono(s[r]) << 32) | (unsigned int)(~(unsigned int)m);
        ins4(key, top[r]);
      }
    }
  }
#else
  // ---- fallback: direct VMEM loads ----
  const int mEnd = chunk * CHROWS + CHROWS;
  for (int m0 = chunk * CHROWS + wave * 16; m0 < mEnd; m0 += 64) {
    const int m = m0 + nlo;
    const float* __restrict__ rowp = base + (size_t)m * CW_;
    if (m + 64 < MEM_) __builtin_prefetch(rowp + 64 * CW_, 0, 0);
    float sq = 0.f;
    float s0, s1, s2, s3;
#if USE_WMMA
    v8f acc = {0.f, 0.f, 0.f, 0.f, 0.f, 0.f, 0.f, 0.f};
    #pragma unroll
    for (int t = 0; t < 16; ++t) {
      const int w = 4 * t + 2 * half;
      float b0 = rowp[w], b1 = rowp[w + 1];
      v2f av; av.x = A0[t]; av.y = A1[t];
      v2f bv; bv.x = b0;    bv.y = b1;
      acc = __builtin_amdgcn_wmma_f32_16x16x4_f32(false, av, false, bv, (short)0, acc, false, false);
      sq += b0 * b0 + b1 * b1;
    }
    s0 = acc[0]; s1 = acc[1]; s2 = acc[2]; s3 = acc[3];
#else
    float d0 = 0.f, d1 = 0.f, d2 = 0.f, d3 = 0.f;
    #pragma unroll
    for (int w0 = 0; w0 < 32; ++w0) {
      const int w = 32 * half + w0;
      float v = rowp[w];
      sq += v * v;
      d0 += qs[0][w] * v; d1 += qs[1][w] * v; d2 += qs[2][w] * v; d3 += qs[3][w] * v;
    }
    d0 += __shfl_xor(d0, 16, 32); d1 += __shfl_xor(d1, 16, 32);
    d2 += __shfl_xor(d2, 16, 32); d3 += __shfl_xor(d3, 16, 32);
    s0 = 2.f * d0; s1 = 2.f * d1; s2 = 2.f * d2; s3 = 2.f * d3;
#endif
    float fsq = sq + __shfl_xor(sq, 16, 32);
    if (half == 0) {
      float s[4] = { s0 - fsq, s1 - fsq, s2 - fsq, s3 - fsq };
      int ov = -1;
      #pragma unroll
      for (int j = 0; j < C_; ++j) if (rps[j] == m) ov = j;
      if (ov >= 0) { s[0] = nscs[ov][0]; s[1] = nscs[ov][1]; s[2] = nscs[ov][2]; s[3] = nscs[ov][3]; }
      #pragma unroll
      for (int r = 0; r < R_; ++r) {
        unsigned long long key =
            ((unsigned long long)fmono(s[r]) << 32) | (unsigned int)(~(unsigned int)m);
        ins4(key, top[r]);
      }
    }
  }
#endif

  // block-level merge
  __shared__ unsigned long long tb[4][16][R_][4];
  if (half == 0)
    for (int r = 0; r < R_; ++r)
      for (int e = 0; e < 4; ++e) tb[wave][nlo][r][e] = top[r][e];
  __syncthreads();
  if (threadIdx.x < R_) {
    const int r = threadIdx.x;
    unsigned long long t4[4] = {0, 0, 0, 0};
    for (int wv = 0; wv < 4; ++wv)
      for (int l = 0; l < 16; ++l)
        for (int e = 0; e < 4; ++e) ins4(tb[wv][l][r][e], t4);
    for (int e = 0; e < 4; ++e) ws->btop[b][chunk][r][e] = t4[e];
  }
}

// ---------------- kernel 4: cross-chunk merge ----------------
__global__ __launch_bounds__(32) void merge_kernel(WS* __restrict__ ws) {
  const int t = threadIdx.x;
  if (t >= B_ * R_) return;
  const int b = t >> 2, r = t & 3;
  unsigned long long t4[4] = {0, 0, 0, 0};
  for (int ch = 0; ch < NCH; ++ch)
    for (int e = 0; e < 4; ++e) ins4(ws->btop[b][ch][r][e], t4);
  for (int e = 0; e < 4; ++e)
    ws->knn[b][r * K_ + e] = (int)(~(unsigned int)(t4[e] & 0xFFFFFFFFull));
}

// ---------------- kernel 5: gather, cosine sim, softmax, read vectors ----------------
__global__ __launch_bounds__(64) void finalize_kernel(const float* __restrict__ mem,
                                                      const int* __restrict__ rp,
                                                      const WS* __restrict__ ws,
                                                      float* __restrict__ out) {
  const int b = blockIdx.x;
  const int w = threadIdx.x;  // 0..63
  __shared__ int pos[C_];
  __shared__ int rps[C_];
  __shared__ float vis[C_][CW_];
  __shared__ float kq[R_][CW_];
  __shared__ float sim[R_][C_];
  __shared__ float vnorm[C_];
  __shared__ float qnorm[R_];
  if (w < C_) rps[w] = rp[b * C_ + w];
  __syncthreads();
  if (w == 0) {
    unsigned int maxlen = (unsigned int)(ws->amin & 0xFFFFFFFFull);  // least_used[0,0]
    for (int i = 0; i < R_ * K_; ++i)  pos[i] = (int)umin_u((unsigned int)ws->knn[b][i], maxlen);
    for (int i = 0; i < 2 * KL_; ++i)  pos[R_ * K_ + i] = (int)umin_u((unsigned int)ws->fpbp[b][i], maxlen);
    pos[C_ - 1] = (int)umin_u((unsigned int)(C_ + 1), maxlen);       // lum = C+1
  }
  __syncthreads();
  for (int c = 0; c < C_; ++c) {
    int p = pos[c];
    int ov = -1;
    for (int j = 0; j < C_; ++j) if (rps[j] == p) ov = j;            // updated memory row?
    vis[c][w] = (ov >= 0) ? ws->vmnew[b][ov][w] : mem[((size_t)b * MEM_ + p) * CW_ + w];
  }
  for (int r = 0; r < R_; ++r) kq[r][w] = ws->rq[b][r][w];
  __syncthreads();
  if (w < R_) { float s = 0.f; for (int i = 0; i < CW_; ++i) s += kq[w][i] * kq[w][i]; qnorm[w] = sqrtf(s) + EPS_; }
  if (w < C_) { float s = 0.f; for (int i = 0; i < CW_; ++i) s += vis[w][i] * vis[w][i]; vnorm[w] = sqrtf(s) + EPS_; }
  __syncthreads();
  for (int i = w; i < R_ * C_; i += 64) {
    int r = i / C_, c = i % C_;
    float s = 0.f;
    for (int k = 0; k < CW_; ++k) s += kq[r][k] * vis[c][k];
    sim[r][c] = s / (qnorm[r] * vnorm[c]);
  }
  __syncthreads();
  if (w < R_) {
    float mx = -1e30f;
    for (int c = 0; c < C_; ++c) mx = fmaxf(mx, sim[w][c]);
    float ssum = 0.f;
    for (int c = 0; c < C_; ++c) { float e = expf(sim[w][c] - mx); sim[w][c] = e; ssum += e; }
    float inv = 1.f / ssum;
    for (int c = 0; c < C_; ++c) sim[w][c] *= inv;
  }
  __syncthreads();
  for (int r = 0; r < R_; ++r) {
    float s = 0.f;
    for (int c = 0; c < C_; ++c) s += sim[r][c] * vis[c][w];
    out[((size_t)b * R_ + r) * CW_ + w] = s;
  }
}

// ---------------- launch ----------------
extern "C" void kernel_launch(void* const* d_in, const int* in_sizes, int n_in,
                              void* d_out, int out_size, void* d_ws, size_t ws_size,
                              hipStream_t stream) {
  (void)in_sizes; (void)n_in; (void)out_size; (void)ws_size;
  const float* xi    = (const float*)d_in[0];
  const float* mem   = (const float*)d_in[1];
  const float* vmem  = (const float*)d_in[2];
  const float* link  = (const float*)d_in[3];
  const float* rev   = (const float*)d_in[4];
  const float* prec  = (const float*)d_in[5];
  const float* rwt   = (const float*)d_in[6];
  const float* wwt   = (const float*)d_in[7];
  const float* usage = (const float*)d_in[8];
  // d_in[9] = least_used_mem: unused by the reference computation
  const int*   rp    = (const int*)d_in[10];
  const float* W_rq  = (const float*)d_in[11];
  const float* b_rq  = (const float*)d_in[12];
  const float* W_wv  = (const float*)d_in[13];
  const float* b_wv  = (const float*)d_in[14];
  const float* W_ig  = (const float*)d_in[15];
  const float* b_ig  = (const float*)d_in[16];
  const float* W_wg  = (const float*)d_in[17];
  const float* b_wg  = (const float*)d_in[18];
  WS* ws = (WS*)d_ws;
  float* out = (float*)d_out;

  prep_kernel<<<dim3(B_), dim3(256), 0, stream>>>(xi, vmem, link, rev, prec, rwt, wwt, usage, rp,
                                                  W_rq, b_rq, W_wv, b_wv, W_ig, b_ig, W_wg, b_wg, ws);
  argmin_kernel<<<dim3(64), dim3(256), 0, stream>>>(usage, rp, ws);
  topk_kernel<<<dim3(NCH, B_), dim3(128), 0, stream>>>(mem, rp, ws);
  merge_kernel<<<dim3(1), dim3(32), 0, stream>>>(ws);
  finalize_kernel<<<dim3(B_), dim3(64), 0, stream>>>(mem, rp, ws, out);
}